// GCNModule_46024869544086
// MI455X (gfx1250) — compile-verified
//
#include <hip/hip_runtime.h>
#include <math.h>

typedef __attribute__((ext_vector_type(2))) float v2f;
typedef __attribute__((ext_vector_type(8))) float v8f;

#define GDIM 128

// ---------------- degree / normalization ----------------
__global__ __launch_bounds__(256) void k_deg_init(float* __restrict__ deg, int n) {
  int i = blockIdx.x * 256 + threadIdx.x;
  if (i < n) deg[i] = 1.0f;  // self-loop
}

__global__ __launch_bounds__(256) void k_deg_count(const int* __restrict__ col,
                                                   float* __restrict__ deg, int ne) {
  int e = blockIdx.x * 256 + threadIdx.x;
  if (e < ne) atomicAdd(&deg[col[e]], 1.0f);
}

__global__ __launch_bounds__(256) void k_deg_rsqrt(float* __restrict__ deg, int n) {
  int i = blockIdx.x * 256 + threadIdx.x;
  if (i < n) deg[i] = rsqrtf(deg[i]);  // deg >= 1 always (self-loop)
}

// ---------------- xw = x @ W via fp32 WMMA ----------------
// One block = one 16-row tile of x; 8 waves cover the 8 column tiles of W.
// A-frag (16x4 f32): lane 0-15 -> M=lane, K={k,k+1}; lane 16-31 -> M=lane-16, K={k+2,k+3}
// B-frag (4x16 f32): N = lane&15; K split identically across half-waves.
// D (16x16 f32, 8 VGPRs): VGPR v -> M = v + 8*(lane>=16), N = lane&15.
__global__ __launch_bounds__(256) void k_gemm_wmma(const float* __restrict__ x,
                                                   const float* __restrict__ W,
                                                   float* __restrict__ xw, int n) {
  __shared__ float xs[16 * 132];  // 16x128 tile, stride 132 floats -> conflict-free column reads
  const int m0 = blockIdx.x * 16;
  if (m0 + 16 > n) return;  // block-uniform: EXEC stays all-ones for WMMA

  // cooperative copy of a contiguous 16x128 fp32 tile into LDS (512 float4 chunks)
  const float* src = x + (size_t)m0 * GDIM;
  for (int t = threadIdx.x; t < 512; t += 256) {
    const int row = t >> 5, seg = t & 31;
    const float4 val = reinterpret_cast<const float4*>(src)[(row << 5) + seg];
    *reinterpret_cast<float4*>(&xs[row * 132 + (seg << 2)]) = val;  // 132 % 4 == 0 -> aligned
  }
  __syncthreads();

  const int lane = threadIdx.x & 31;
  const int n0   = (threadIdx.x >> 5) << 4;  // wave-id * 16
  const int r    = lane & 15;
  const int kh   = (lane >> 4) << 1;         // 0 for lanes 0-15, 2 for lanes 16-31
  const int ncol = n0 + r;

  v8f c = {};
#pragma unroll
  for (int k = 0; k < GDIM; k += 4) {
    v2f a, b;
    a.x = xs[r * 132 + k + kh];
    a.y = xs[r * 132 + k + kh + 1];
    b.x = W[(size_t)(k + kh) * GDIM + ncol];
    b.y = W[(size_t)(k + kh + 1) * GDIM + ncol];
    // 8 args: (neg_a, A, neg_b, B, c_mod, C, reuse_a, reuse_b)
    c = __builtin_amdgcn_wmma_f32_16x16x4_f32(false, a, false, b, (short)0, c,
                                              false, false);
  }

  const int mrow = m0 + ((lane >> 4) << 3);
  float* o = xw + (size_t)mrow * GDIM + ncol;
#pragma unroll
  for (int vv = 0; vv < 8; ++vv) o[(size_t)vv * GDIM] = c[vv];
}

// ---------------- out[i,:] = dinv[i]^2 * xw[i,:]  (self-loop term, full init) ----------------
__global__ __launch_bounds__(256) void k_selfloop(const float* __restrict__ dinv,
                                                  const float* __restrict__ xw,
                                                  float* __restrict__ out, int n) {
  int i = blockIdx.x * 256 + threadIdx.x;  // over n*32 float4 chunks
  if (i >= n * 32) return;
  const int node = i >> 5, q = i & 31;
  float s = dinv[node];
  s = s * s;
  const float4 v = reinterpret_cast<const float4*>(xw + (size_t)node * GDIM)[q];
  float4 o;
  o.x = s * v.x; o.y = s * v.y; o.z = s * v.z; o.w = s * v.w;
  reinterpret_cast<float4*>(out + (size_t)node * GDIM)[q] = o;
}

// ---------------- per-edge gather + scatter-add (one wave32 per edge) ----------------
__global__ __launch_bounds__(256) void k_edge_scatter(const int* __restrict__ row,
                                                      const int* __restrict__ col,
                                                      const float* __restrict__ dinv,
                                                      const float* __restrict__ xw,
                                                      float* __restrict__ out, int ne) {
  const int e = (blockIdx.x * 256 + threadIdx.x) >> 5;
  if (e >= ne) return;
  const int lane = threadIdx.x & 31;
  const int s = row[e];  // source (gather)
  const int d = col[e];  // destination (scatter)
  const float norm = dinv[s] * dinv[d];
  const float4 v = reinterpret_cast<const float4*>(xw + (size_t)s * GDIM)[lane];
  float* o = out + (size_t)d * GDIM + (lane << 2);
  atomicAdd(o + 0, norm * v.x);
  atomicAdd(o + 1, norm * v.y);
  atomicAdd(o + 2, norm * v.z);
  atomicAdd(o + 3, norm * v.w);
}

// ---------------- out = gelu_exact(out + b) ----------------
__global__ __launch_bounds__(256) void k_bias_gelu(float* __restrict__ out,
                                                   const float* __restrict__ b, int n) {
  int i = blockIdx.x * 256 + threadIdx.x;  // over n*32 float4 chunks
  if (i >= n * 32) return;
  const float4 bb = reinterpret_cast<const float4*>(b)[i & 31];
  float4 v = reinterpret_cast<float4*>(out)[i];
  float t[4] = {v.x + bb.x, v.y + bb.y, v.z + bb.z, v.w + bb.w};
#pragma unroll
  for (int j = 0; j < 4; ++j) {
    const float u = t[j];
    t[j] = 0.5f * u * (1.0f + erff(u * 0.70710678118654752440f));
  }
  v.x = t[0]; v.y = t[1]; v.z = t[2]; v.w = t[3];
  reinterpret_cast<float4*>(out)[i] = v;
}

extern "C" void kernel_launch(void* const* d_in, const int* in_sizes, int n_in,
                              void* d_out, int out_size, void* d_ws, size_t ws_size,
                              hipStream_t stream) {
  const float* x  = (const float*)d_in[0];
  const int*   ei = (const int*)d_in[1];
  const float* W  = (const float*)d_in[2];
  const float* b  = (const float*)d_in[3];
  float*       out = (float*)d_out;

  const int n  = in_sizes[0] / GDIM;  // 100000
  const int ne = in_sizes[1] / 2;     // 600000
  const int* erow = ei;               // edge_index[0] = sources
  const int* ecol = ei + ne;          // edge_index[1] = destinations

  // workspace: dinv[n] floats, then xw[n*128] floats (~52 MB total)
  float* dinv = (float*)d_ws;
  const size_t off = ((size_t)n + 1023) & ~(size_t)1023;
  float* xw = dinv + off;

  const int bn  = (n + 255) / 256;
  const int bne = (ne + 255) / 256;
  const int bv  = (n * 32 + 255) / 256;          // n*128 elems as float4
  const int bedge = ((ne * 32) + 255) / 256;     // one wave per edge
  const int btile = n / 16;                      // n is a multiple of 16

  k_deg_init   <<<bn,    256, 0, stream>>>(dinv, n);
  k_deg_count  <<<bne,   256, 0, stream>>>(ecol, dinv, ne);
  k_deg_rsqrt  <<<bn,    256, 0, stream>>>(dinv, n);
  k_gemm_wmma  <<<btile, 256, 0, stream>>>(x, W, xw, n);
  k_selfloop   <<<bv,    256, 0, stream>>>(dinv, xw, out, n);
  k_edge_scatter<<<bedge,256, 0, stream>>>(erow, ecol, dinv, xw, out, ne);
  k_bias_gelu  <<<bv,    256, 0, stream>>>(out, b, n);
}